// QuantumGenerator_10213432230564
// MI455X (gfx1250) — compile-verified
//
#include <hip/hip_runtime.h>

#define NQ       14
#define NSTATE   16384     // 2^14
#define NTHREADS 256       // 8 waves of 32
#define NLAYERS  3

typedef __attribute__((ext_vector_type(2))) float v2f;
typedef __attribute__((ext_vector_type(8))) float v8f;

// V_WMMA_F32_16X16X4_F32 : D(16x16,f32) = A(16x4,f32) x B(4x16,f32) + C
__device__ __forceinline__ v8f wmma4(v2f a, v2f b, v8f c) {
  return __builtin_amdgcn_wmma_f32_16x16x4_f32(false, a, false, b, (short)0, c,
                                               false, false);
}

// CNOT as in-place LDS permutation: for indices with control bit=1, target
// bit=0, swap with (index | target_bit). Pairs are disjoint within one CNOT.
__device__ __forceinline__ void cnot_swap(float* sRe, float* sIm,
                                          int bc, int bt, int tid) {
  const int hi = bc > bt ? bc : bt;
  const int lo = bc > bt ? bt : bc;
  for (int k = tid; k < (NSTATE >> 2); k += NTHREADS) {
    unsigned low  = (unsigned)k & ((1u << lo) - 1u);
    unsigned mid  = ((unsigned)k >> lo) & ((1u << (hi - lo - 1)) - 1u);
    unsigned high = (unsigned)k >> (hi - 1);
    unsigned idx  = (high << (hi + 1)) | (mid << (lo + 1)) | low | (1u << bc);
    unsigned par  = idx | (1u << bt);
    float tr = sRe[idx], ti = sIm[idx];
    sRe[idx] = sRe[par]; sIm[idx] = sIm[par];
    sRe[par] = tr;       sIm[par] = ti;
  }
}

__global__ __launch_bounds__(NTHREADS)
void qgen_kernel(const float* __restrict__ z_batch,
                 const float* __restrict__ theta,
                 float* __restrict__ out)
{
  // Statevector resident in LDS (split re/im): 128 KB of the 320 KB WGP LDS.
  __shared__ float sRe[NSTATE];
  __shared__ float sIm[NSTATE];
  __shared__ float gRe[NQ][4],  gIm[NQ][4];          // fused per-wire 2x2 gates
  __shared__ float uRe[3][16][16], uIm[3][16][16];   // 16x16 group unitaries
  __shared__ float u4Re[4][4], u4Im[4][4];           // wires 12,13 (4x4)
  __shared__ float wsum[8][NQ];                      // per-wave reduction

  const int tid  = threadIdx.x;
  const int b    = blockIdx.x;
  const int lane = tid & 31;
  const int wv   = tid >> 5;

  // |00...0>
  for (int i = tid; i < NSTATE; i += NTHREADS) { sRe[i] = 0.f; sIm[i] = 0.f; }
  if (tid == 0) sRe[0] = 1.f;
  __syncthreads();

  // stage -1 = latent encoding (RZ*RY per wire), stages 0..2 = variational
  for (int stage = -1; stage < NLAYERS; ++stage) {
    // ---- build fused per-wire 2x2 gates ----
    if (tid < NQ) {
      int w = tid;
      float g00r,g00i,g01r,g01i,g10r,g10i,g11r,g11i;
      if (stage < 0) {
        float t = z_batch[b*NQ + w];
        float s, c; __sincosf(0.5f*t, &s, &c);
        // G = RZ(t)*RY(t):  row0 *= exp(-it/2), row1 *= exp(+it/2)
        g00r =  c*c; g00i = -s*c;
        g01r = -c*s; g01i =  s*s;
        g10r =  c*s; g10i =  s*s;
        g11r =  c*c; g11i =  s*c;
      } else {
        const float* th = theta + (stage*NQ + w)*3;
        float s0,c0,s1,c1,s2,c2;
        __sincosf(0.5f*th[0], &s0, &c0);
        __sincosf(0.5f*th[1], &s1, &c1);
        __sincosf(0.5f*th[2], &s2, &c2);
        // M = RY(th1)*RX(th0)
        float M00r =  c1*c0, M00i =  s1*s0;
        float M01r = -s1*c0, M01i = -c1*s0;
        float M10r =  s1*c0, M10i = -c1*s0;
        float M11r =  c1*c0, M11i = -s1*s0;
        // G = RZ(th2)*M : row0 *= (c2 - i s2), row1 *= (c2 + i s2)
        g00r = c2*M00r + s2*M00i; g00i = c2*M00i - s2*M00r;
        g01r = c2*M01r + s2*M01i; g01i = c2*M01i - s2*M01r;
        g10r = c2*M10r - s2*M10i; g10i = c2*M10i + s2*M10r;
        g11r = c2*M11r - s2*M11i; g11i = c2*M11i + s2*M11r;
      }
      gRe[w][0]=g00r; gIm[w][0]=g00i;
      gRe[w][1]=g01r; gIm[w][1]=g01i;
      gRe[w][2]=g10r; gIm[w][2]=g10i;
      gRe[w][3]=g11r; gIm[w][3]=g11i;
    }
    __syncthreads();

    // ---- tensor products: U_g = G_{4g} (x) G_{4g+1} (x) G_{4g+2} (x) G_{4g+3}
    {
      int i = tid >> 4, j = tid & 15;   // one 16x16 entry per thread per group
      for (int g = 0; g < 3; ++g) {
        float re = 1.f, im = 0.f;
        #pragma unroll
        for (int q = 0; q < 4; ++q) {   // wire 4g+q owns bit (3-q) of i/j
          int ib = (i >> (3 - q)) & 1;
          int jb = (j >> (3 - q)) & 1;
          float ar = gRe[4*g + q][ib*2 + jb];
          float ai = gIm[4*g + q][ib*2 + jb];
          float nr = re*ar - im*ai;
          float ni = re*ai + im*ar;
          re = nr; im = ni;
        }
        uRe[g][i][j] = re; uIm[g][i][j] = im;
      }
      if (tid < 16) {                   // wires 12,13 -> 4x4
        int i4 = tid >> 2, j4 = tid & 3;
        float ar = gRe[12][((i4>>1)&1)*2 + ((j4>>1)&1)];
        float ai = gIm[12][((i4>>1)&1)*2 + ((j4>>1)&1)];
        float br = gRe[13][(i4&1)*2 + (j4&1)];
        float bi = gIm[13][(i4&1)*2 + (j4&1)];
        u4Re[i4][j4] = ar*br - ai*bi;
        u4Im[i4][j4] = ar*bi + ai*br;
      }
    }
    __syncthreads();

    // ---- apply each 16x16 group unitary as complex GEMM U(16x16)xS(16x1024)
    // via V_WMMA_F32_16X16X4_F32 (K=16 as 4 chained K=4 WMMAs).
    // 64 column tiles of 16; 8 waves x 8 tiles each.
    for (int g = 0; g < 3; ++g) {
      const int cl = 10 - 4*g;            // log2(C), C = stride below the group
      const int m  = lane & 15;
      const int kh = lane >> 4;
      // A fragments (ISA layout: VGPR0 K={0,2} halves, VGPR1 K={1,3}):
      v2f aR[4], aI[4], aN[4];
      #pragma unroll
      for (int k = 0; k < 4; ++k) {
        int kb = 4*k + 2*kh;
        aR[k].x =  uRe[g][m][kb];   aR[k].y =  uRe[g][m][kb+1];
        aI[k].x =  uIm[g][m][kb];   aI[k].y =  uIm[g][m][kb+1];
        aN[k].x = -aI[k].x;         aN[k].y = -aI[k].y;   // F32 WMMA has no A-neg
      }
      for (int t8 = 0; t8 < 8; ++t8) {
        const int col   = (wv*8 + t8)*16 + m;
        const int cmask = (1 << cl) - 1;
        const int base  = ((col >> cl) << (cl + 4)) | (col & cmask); // + j<<cl
        v2f bR[4], bI[4];
        #pragma unroll
        for (int k = 0; k < 4; ++k) {
          int kb = 4*k + 2*kh;
          int i0 = base + (kb << cl);
          int i1 = base + ((kb + 1) << cl);
          bR[k].x = sRe[i0]; bR[k].y = sRe[i1];
          bI[k].x = sIm[i0]; bI[k].y = sIm[i1];
        }
        v8f accR = {0.f,0.f,0.f,0.f,0.f,0.f,0.f,0.f};
        v8f accI = {0.f,0.f,0.f,0.f,0.f,0.f,0.f,0.f};
        #pragma unroll
        for (int k = 0; k < 4; ++k) accR = wmma4(aR[k], bR[k], accR); //  Ur*Sr
        #pragma unroll
        for (int k = 0; k < 4; ++k) accR = wmma4(aN[k], bI[k], accR); // -Ui*Si
        #pragma unroll
        for (int k = 0; k < 4; ++k) accI = wmma4(aR[k], bI[k], accI); //  Ur*Si
        #pragma unroll
        for (int k = 0; k < 4; ++k) accI = wmma4(aI[k], bR[k], accI); //  Ui*Sr
        // D layout: VGPR v -> row v + 8*kh, col = m (same columns we read)
        #pragma unroll
        for (int v = 0; v < 8; ++v) {
          int idx = base + ((v + 8*kh) << cl);
          sRe[idx] = accR[v];
          sIm[idx] = accI[v];
        }
      }
      __syncthreads();
    }

    // ---- wires 12,13: 4x4 complex unitary on state viewed as (4096,4) ----
    for (int r = 0; r < 16; ++r) {
      int base = (r*NTHREADS + tid) * 4;
      float xr[4], xi[4];
      #pragma unroll
      for (int j = 0; j < 4; ++j) { xr[j] = sRe[base+j]; xi[j] = sIm[base+j]; }
      #pragma unroll
      for (int i = 0; i < 4; ++i) {
        float yr = 0.f, yi = 0.f;
        #pragma unroll
        for (int j = 0; j < 4; ++j) {
          yr += u4Re[i][j]*xr[j] - u4Im[i][j]*xi[j];
          yi += u4Re[i][j]*xi[j] + u4Im[i][j]*xr[j];
        }
        sRe[base+i] = yr; sIm[base+i] = yi;
      }
    }
    __syncthreads();

    // ---- CNOT ring (variational layers only); wire w owns bit (13-w) ----
    if (stage >= 0) {
      for (int w = 0; w < NQ - 1; ++w) {
        cnot_swap(sRe, sIm, 13 - w, 12 - w, tid);
        __syncthreads();
      }
      cnot_swap(sRe, sIm, 0, 13, tid);   // CNOT(control=wire13, target=wire0)
      __syncthreads();
    }
  }

  // ---- <Z_w> = sum_i (-1)^{bit_{13-w}(i)} |amp_i|^2 ----
  float ev[NQ];
  #pragma unroll
  for (int w = 0; w < NQ; ++w) ev[w] = 0.f;
  for (int r = 0; r < NSTATE/NTHREADS; ++r) {
    int i = r*NTHREADS + tid;
    float p = sRe[i]*sRe[i] + sIm[i]*sIm[i];
    #pragma unroll
    for (int w = 0; w < NQ; ++w)
      ev[w] += ((i >> (13 - w)) & 1) ? -p : p;
  }
  #pragma unroll
  for (int w = 0; w < NQ; ++w) {
    float v = ev[w];
    for (int off = 16; off > 0; off >>= 1) v += __shfl_down(v, off, 32);
    if (lane == 0) wsum[wv][w] = v;
  }
  __syncthreads();
  if (tid < NQ) {
    float s = 0.f;
    #pragma unroll
    for (int q = 0; q < 8; ++q) s += wsum[q][tid];
    out[b*NQ + tid] = s;   // SCALE == 1.0
  }
}

extern "C" void kernel_launch(void* const* d_in, const int* in_sizes, int n_in,
                              void* d_out, int out_size, void* d_ws, size_t ws_size,
                              hipStream_t stream) {
  (void)in_sizes; (void)n_in; (void)d_ws; (void)ws_size; (void)out_size;
  const float* z_batch = (const float*)d_in[0];   // (256, 14) f32
  const float* theta   = (const float*)d_in[1];   // (3, 14, 3) f32
  float* out = (float*)d_out;                     // (256, 14) f32
  qgen_kernel<<<dim3(256), dim3(NTHREADS), 0, stream>>>(z_batch, theta, out);
}